// Model_16527034155660
// MI455X (gfx1250) — compile-verified
//
#include <hip/hip_runtime.h>
#include <math.h>

typedef __attribute__((ext_vector_type(2))) float v2f;
typedef __attribute__((ext_vector_type(8))) float v8f;

#define BATCH 32
#define SEQ   4096
#define DIM   768
#define HEADS 8
#define KEEP  2048

// workspace layout in floats
#define WS_U      768                       // padded U: [DIM][16] = 12288 floats
#define WS_DOTS   (768 + 12288)             // dots: [B][H][L] = 1048576 floats
#define WS_SCORE  (WS_DOTS + BATCH*HEADS*SEQ)   // score: [B][L] = 131072 floats
#define WS_IDS    (WS_SCORE + BATCH*SEQ)        // ids: [B][KEEP] ints

// ---------------------------------------------------------------------------
// Kernel 1: q = score_emb @ Wq^T ; U[k][h] = sum_{j in head h} q[j]*Wc[j][k]
// One block, 768 threads. Pads heads 8..15 with zeros for the 16-wide WMMA N.
// ---------------------------------------------------------------------------
__global__ void proj_kernel(const float* __restrict__ se,
                            const float* __restrict__ wq,
                            const float* __restrict__ wc,
                            float* __restrict__ u) {
    __shared__ float s_q[DIM];
    const int j = threadIdx.x;           // 0..767
    float acc = 0.f;
    for (int i = 0; i < DIM; ++i) acc += se[i] * wq[(size_t)j * DIM + i];
    s_q[j] = acc;
    __syncthreads();

    const int k = threadIdx.x;           // column of Wc
    for (int h = 0; h < 16; ++h) {
        float a = 0.f;
        if (h < HEADS) {
            const int j0 = h * (DIM / HEADS);
            for (int jj = 0; jj < DIM / HEADS; ++jj)
                a += s_q[j0 + jj] * wc[(size_t)(j0 + jj) * DIM + k];
        }
        u[(size_t)k * 16 + h] = a;       // U stored as [DIM][16] (K-major, N inner)
    }
}

// ---------------------------------------------------------------------------
// Kernel 2: dots[b,h,l] = scale * <x[b,l,:], U[:,h]>  via V_WMMA_F32_16X16X4_F32
// Each wave: one 16-token x 16-col tile, K-loop over DIM in steps of 4.
// A layout: lane m=lane&15, VGPR v -> K = kk + 2*(lane>>4) + v  (one b64 load)
// B layout: lane n=lane&15, VGPR v -> K = kk + 2*(lane>>4) + v  (from LDS)
// C layout: VGPR v, lane -> (M = v + 8*(lane>>4), N = lane&15)
// ---------------------------------------------------------------------------
__global__ void score_wmma_kernel(const float* __restrict__ x,
                                  const float* __restrict__ u,
                                  float* __restrict__ dots) {
    __shared__ float s_u[DIM * 16];      // 48 KB
    for (int i = threadIdx.x; i < DIM * 16; i += blockDim.x) s_u[i] = u[i];
    __syncthreads();

    const int lane = threadIdx.x & 31;
    const int wave = threadIdx.x >> 5;
    const int tile = blockIdx.x * 8 + wave;   // 8192 tiles total
    const int t0   = tile * 16;               // flat token base (never crosses batch)
    const int m    = lane & 15;
    const int n    = lane & 15;
    const int koff = (lane >> 4) * 2;

    const float* xrow = x + (size_t)(t0 + m) * DIM + koff;
    const float* brow = s_u + (size_t)koff * 16 + n;

    v8f c = {};
#pragma unroll 4
    for (int kk = 0; kk < DIM; kk += 4) {
        v2f a = *(const v2f*)(xrow + kk);
        v2f b;
        b.x = brow[(kk + 0) * 16];
        b.y = brow[(kk + 1) * 16];
        c = __builtin_amdgcn_wmma_f32_16x16x4_f32(
                /*neg_a=*/false, a, /*neg_b=*/false, b,
                /*c_mod=*/(short)0, c, /*reuse_a=*/false, /*reuse_b=*/false);
    }

    const float scale = 0.10206207261596577f;   // 96^-0.5
    const int b_idx = t0 >> 12;                 // t0 / SEQ
    const int tok   = t0 & (SEQ - 1);
    if (n < HEADS) {
        float* drow = dots + ((size_t)(b_idx * HEADS + n)) * SEQ
                           + tok + (lane >> 4) * 8;
#pragma unroll
        for (int v = 0; v < 8; ++v) drow[v] = c[v] * scale;
    }
}

// ---------------------------------------------------------------------------
// Kernel 3: per (b,h) softmax over L, mean over heads -> score[b][l]
// One block per batch, 256 threads; each thread owns 16 consecutive l.
// ---------------------------------------------------------------------------
__global__ void softmax_mean_kernel(const float* __restrict__ dots,
                                    float* __restrict__ score) {
    const int b   = blockIdx.x;
    const int tid = threadIdx.x;      // 256
    __shared__ float red[256];

    float acc[16];
#pragma unroll
    for (int i = 0; i < 16; ++i) acc[i] = 0.f;

    for (int h = 0; h < HEADS; ++h) {
        const float* dp = dots + ((size_t)(b * HEADS + h)) * SEQ + tid * 16;
        float v[16];
#pragma unroll
        for (int i = 0; i < 16; ++i) v[i] = dp[i];

        float m = v[0];
#pragma unroll
        for (int i = 1; i < 16; ++i) m = fmaxf(m, v[i]);
        red[tid] = m; __syncthreads();
        for (int s = 128; s > 0; s >>= 1) {
            if (tid < s) red[tid] = fmaxf(red[tid], red[tid + s]);
            __syncthreads();
        }
        const float gmax = red[0]; __syncthreads();

        float e[16]; float ls = 0.f;
#pragma unroll
        for (int i = 0; i < 16; ++i) { e[i] = expf(v[i] - gmax); ls += e[i]; }
        red[tid] = ls; __syncthreads();
        for (int s = 128; s > 0; s >>= 1) {
            if (tid < s) red[tid] += red[tid + s];
            __syncthreads();
        }
        const float inv = 1.0f / red[0]; __syncthreads();
#pragma unroll
        for (int i = 0; i < 16; ++i) acc[i] += e[i] * inv;
    }

    float* sp = score + (size_t)b * SEQ + tid * 16;
#pragma unroll
    for (int i = 0; i < 16; ++i) sp[i] = acc[i] * (1.0f / HEADS);
}

// ---------------------------------------------------------------------------
// Kernel 4: per-batch descending argsort via in-LDS bitonic sort of packed
// 64-bit keys ((ordered_f32<<32) | (4095 - l)); stable w.r.t. smaller index.
// ---------------------------------------------------------------------------
__global__ void topk_sort_kernel(const float* __restrict__ score,
                                 int* __restrict__ ids,
                                 float* __restrict__ out_ids) {
    const int b   = blockIdx.x;
    const int tid = threadIdx.x;   // 1024
    __shared__ unsigned long long s[SEQ];   // 32 KB

    for (int l = tid; l < SEQ; l += 1024) {
        const unsigned int u = __float_as_uint(score[(size_t)b * SEQ + l]);
        const unsigned int mask = (unsigned int)(((int)u) >> 31) | 0x80000000u;
        const unsigned int ord = u ^ mask;     // ascending uint == ascending float
        s[l] = ((unsigned long long)ord << 32) | (unsigned int)(SEQ - 1 - l);
    }
    __syncthreads();

    for (int k = 2; k <= SEQ; k <<= 1) {
        for (int j = k >> 1; j > 0; j >>= 1) {
            for (int i = tid; i < SEQ; i += 1024) {
                const int ixj = i ^ j;
                if (ixj > i) {
                    const unsigned long long a = s[i];
                    const unsigned long long c = s[ixj];
                    const bool up = ((i & k) == 0);
                    if ((a > c) == up) { s[i] = c; s[ixj] = a; }
                }
            }
            __syncthreads();
        }
    }
    // ascending array -> read from the top for descending order
    for (int r = tid; r < KEEP; r += 1024) {
        const unsigned long long key = s[SEQ - 1 - r];
        const int l = (SEQ - 1) - (int)(unsigned int)(key & 0xFFFFFFFFu);
        ids[b * KEEP + r] = l;
        out_ids[b * KEEP + r] = (float)l;
    }
}

// ---------------------------------------------------------------------------
// Kernel 5: gather kept rows of x_lead and x_pos_masked (float4, 192 thr/row)
// ---------------------------------------------------------------------------
__global__ void gather_kernel(const float* __restrict__ x_lead,
                              const float* __restrict__ x_pos,
                              const int* __restrict__ ids,
                              float* __restrict__ out) {
    const int r = blockIdx.x;      // 0..KEEP-1
    const int b = blockIdx.y;      // 0..BATCH-1
    const int t = threadIdx.x;     // 0..191  (192 * 4 = 768 floats)
    const int id = ids[b * KEEP + r];
    const size_t src = ((size_t)b * SEQ + id) * DIM;
    const size_t dst = ((size_t)b * KEEP + r) * DIM;
    const float4 v1 = ((const float4*)(x_lead + src))[t];
    const float4 v2 = ((const float4*)(x_pos + src))[t];
    ((float4*)(out + dst))[t] = v1;
    ((float4*)(out + (size_t)BATCH * KEEP * DIM + dst))[t] = v2;
}

// ---------------------------------------------------------------------------
extern "C" void kernel_launch(void* const* d_in, const int* in_sizes, int n_in,
                              void* d_out, int out_size, void* d_ws, size_t ws_size,
                              hipStream_t stream) {
    const float* x_lead = (const float*)d_in[0];
    const float* x_pos  = (const float*)d_in[1];
    const float* se     = (const float*)d_in[2];
    const float* wq     = (const float*)d_in[3];
    const float* wc     = (const float*)d_in[4];

    float* out  = (float*)d_out;
    float* ws   = (float*)d_ws;
    float* u    = ws + WS_U;
    float* dots = ws + WS_DOTS;
    float* score= ws + WS_SCORE;
    int*   ids  = (int*)(ws + WS_IDS);
    float* out_ids = out + (size_t)2 * BATCH * KEEP * DIM;

    proj_kernel<<<1, DIM, 0, stream>>>(se, wq, wc, u);
    score_wmma_kernel<<<(BATCH * SEQ / 16) / 8, 256, 0, stream>>>(x_lead, u, dots);
    softmax_mean_kernel<<<BATCH, 256, 0, stream>>>(dots, score);
    topk_sort_kernel<<<BATCH, 1024, 0, stream>>>(score, ids, out_ids);
    dim3 ggrid(KEEP, BATCH);
    gather_kernel<<<ggrid, 192, 0, stream>>>(x_lead, x_pos, ids, out);
}